// HardBootstrappingLoss1_59992103190992
// MI455X (gfx1250) — compile-verified
//
#include <hip/hip_runtime.h>
#include <hip/hip_bf16.h>
#include <float.h>

// HardBootstrappingLoss for MI455X (gfx1250).
// Memory-bound streaming reduction: 105MB read of y_pred at 23.3TB/s ~ 5us.
// One wave32 per row; async global->LDS double-buffered row staging (CDNA5
// GLOBAL_LOAD_ASYNC_TO_LDS path, tracked with ASYNCcnt). Top-7 selection uses
// value-only shfl_xor butterflies + wave32 ballot masking (5 ds ops + 1 v_cmp
// per selection instead of 10 ds ops). Deterministic two-level scalar
// reductions (no atomics).

#define C_DIM 100
#define NWAVES 4
#define ROWS_PER_WAVE 8
#define ROWS_PER_BLOCK (NWAVES * ROWS_PER_WAVE)
#define RAT_AVG 0.02f

// Low 32 bits of a flat pointer to LDS == workgroup-relative LDS byte address
// (ISA 10.2: LDS aperture maps flat addr[31:0] -> LDS address).
__device__ __forceinline__ unsigned lds_addr32(const void* p) {
    return (unsigned)(unsigned long long)p;
}

// One row = 100 floats = 400B = 25 lanes x b128. One instruction -> ASYNCcnt+1.
__device__ __forceinline__ void async_row_prefetch(const float* __restrict__ grow,
                                                   unsigned lds_base, int lane) {
    if (lane < 25) {
        unsigned voff = (unsigned)lane * 16u;
        asm volatile("global_load_async_to_lds_b128 %0, %1, %2"
                     :
                     : "v"(lds_base + voff), "v"(voff), "s"(grow)
                     : "memory");
    }
}

__device__ __forceinline__ void wait_async_le1() {
    asm volatile("s_wait_asynccnt 0x1" ::: "memory");
}
__device__ __forceinline__ void wait_async_le0() {
    asm volatile("s_wait_asynccnt 0x0" ::: "memory");
}
// Drain outstanding ds_loads before the async engine may overwrite their
// source buffer (no cross-type ordering guarantee between DS and async LDS
// writes). Near-free: results were already consumed.
__device__ __forceinline__ void wait_ds0() {
    asm volatile("s_wait_dscnt 0x0" ::: "memory");
}

__device__ __forceinline__ unsigned wave_ballot32(bool p) {
#if __has_builtin(__builtin_amdgcn_ballot_w32)
    return __builtin_amdgcn_ballot_w32(p);
#else
    return (unsigned)__ballot(p);
#endif
}

__global__ void __launch_bounds__(NWAVES * 32)
hb_loss_main(const float* __restrict__ y_pred,
             const int* __restrict__ y,
             const int* __restrict__ y_cs,
             const int* __restrict__ epoch_ptr,
             float* __restrict__ out,
             float* __restrict__ ws,
             int N, int nblocks) {
    __shared__ alignas(16) float buf[NWAVES][2][C_DIM]; // double-buffered rows
    __shared__ float ostage[8][ROWS_PER_BLOCK];         // per-row outputs
    __shared__ float pstage[NWAVES][4];                 // per-wave partial sums

    const int tid  = threadIdx.x;
    const int w    = tid >> 5;
    const int lane = tid & 31;
    const int epoch = epoch_ptr[0];
    const long long base = (long long)blockIdx.x * ROWS_PER_BLOCK
                         + (long long)w * ROWS_PER_WAVE;

    float loss_s = 0.f, fycs_s = 0.f, tk_s = 0.f, tk2_s = 0.f;

    if (base < N)
        async_row_prefetch(y_pred + base * C_DIM, lds_addr32(&buf[w][0][0]), lane);

    for (int i = 0; i < ROWS_PER_WAVE; ++i) {
        const long long row = base + i;
        const int cur = i & 1;
        const long long nrow = row + 1;
        const bool havenext = (i + 1 < ROWS_PER_WAVE) && (nrow < N);
        if (havenext) {
            wait_ds0();  // prior iteration's ds reads of buf[w][cur^1] done
            async_row_prefetch(y_pred + nrow * C_DIM,
                               lds_addr32(&buf[w][cur ^ 1][0]), lane);
            wait_async_le1();  // async loads complete in order: row i is ready
        } else {
            wait_async_le0();
        }
        if (row >= N) continue;

        const float* x = &buf[w][cur][0];
        const float o0 = x[lane];
        const float o1 = x[lane + 32];
        const float o2 = x[lane + 64];
        const float o3 = (lane < 4) ? x[lane + 96] : -FLT_MAX;

        // ---- top-7 values + argmax: value-only butterflies + ballot mask ----
        float w0 = o0, w1 = o1, w2 = o2, w3 = o3;
        float m[7];
        int g1 = 0;
        #pragma unroll
        for (int k = 0; k < 7; ++k) {
            float bv = fmaxf(fmaxf(w0, w1), fmaxf(w2, w3));
            #pragma unroll
            for (int off = 16; off; off >>= 1)   // wave32: 5 steps
                bv = fmaxf(bv, __shfl_xor(bv, off, 32));
            m[k] = bv;
            // Exactly-one masking: lowest lane holding bv clears its first
            // matching slot (exact even if duplicate values exist).
            const bool has = (w0 == bv) || (w1 == bv) || (w2 == bv) || (w3 == bv);
            const unsigned ball = wave_ballot32(has);
            const int first = __ffs(ball) - 1;
            int slotidx = 0;
            if (lane == first) {
                if      (w0 == bv) { w0 = -FLT_MAX; slotidx = lane; }
                else if (w1 == bv) { w1 = -FLT_MAX; slotidx = lane + 32; }
                else if (w2 == bv) { w2 = -FLT_MAX; slotidx = lane + 64; }
                else               { w3 = -FLT_MAX; slotidx = lane + 96; }
            }
            if (k == 0) g1 = __shfl(slotidx, first, 32); // argmax index
        }
        const float m1v = m[0];

        // ---- softmax denominator (stable) ----
        float zp = expf(o0 - m1v) + expf(o1 - m1v) + expf(o2 - m1v)
                 + ((lane < 4) ? expf(o3 - m1v) : 0.f);
        #pragma unroll
        for (int off = 16; off; off >>= 1) zp += __shfl_xor(zp, off, 32);

        // ---- min logit -> smallest softmax (max_10) ----
        float mn = fminf(fminf(o0, o1), fminf(o2, (lane < 4) ? o3 : FLT_MAX));
        #pragma unroll
        for (int off = 16; off; off >>= 1) mn = fminf(mn, __shfl_xor(mn, off, 32));

        const float invZ = 1.0f / zp;
        const float p1 = invZ;                    // exp(0)/Z
        const float p2 = expf(m[1] - m1v) * invZ;
        const float p3 = expf(m[2] - m1v) * invZ;
        const float p4 = expf(m[3] - m1v) * invZ;
        const float p5 = expf(m[4] - m1v) * invZ;
        const float p7 = expf(m[6] - m1v) * invZ;
        const float pmin = expf(mn - m1v) * invZ;

        const int yi  = y[row];
        const int yci = y_cs[row];
        const float fy   = expf(x[yi]  - m1v) * invZ;
        const float fycs = expf(x[yci] - m1v) * invZ;

        const int s  = (fy < RAT_AVG) ? 1 : 0;
        const int z1 = s ? g1 : yi;
        const int cond  = (s && (p1 / p2 > p2 / pmin) && (fy < p3)) ? 1 : 0;
        const int agree = (!s && (g1 == yi)) ? 1 : 0;

        tk_s   += (g1 == yci) ? (float)(cond + agree) : 0.f;
        tk2_s  += (float)(cond + agree);
        fycs_s += (fycs <= p7) ? 1.f : 0.f;

        const float logZ = logf(zp);
        const float ce = -(x[z1] - m1v - logZ);
        float boot;
        if (epoch > 2000) {
            boot = ce * p1;
        } else {
            int yps = (p1 < 1.0f) ? 1 : 0;
            int ylocal = 1 - s * yps;
            if (epoch > 50) {
                yps = (p1 < 0.95f) ? 1 : 0;
                ylocal = 1 - s * yps - ((g1 != yi) ? (1 - s) : 0);
            }
            boot = ce * (float)ylocal;
        }
        loss_s += boot;

        if (lane == 0) {
            const int l = w * ROWS_PER_WAVE + i;
            ostage[0][l] = p1;
            ostage[1][l] = p2;
            ostage[2][l] = p3;
            ostage[3][l] = p4;
            ostage[4][l] = p5;
            ostage[5][l] = (float)g1;
            ostage[6][l] = (float)z1;
            ostage[7][l] = (float)s;
        }
    }

    if (lane == 0) {
        pstage[w][0] = loss_s;
        pstage[w][1] = fycs_s;
        pstage[w][2] = tk_s;
        pstage[w][3] = tk2_s;
    }
    __syncthreads();

    // Coalesced per-row output stores.
    if (tid < ROWS_PER_BLOCK) {
        const long long r = (long long)blockIdx.x * ROWS_PER_BLOCK + tid;
        if (r < N) {
            const size_t NN = (size_t)N;
            out[2 + 0 * NN + r] = ostage[0][tid]; // max_1
            out[2 + 1 * NN + r] = ostage[1][tid]; // max_2
            out[2 + 2 * NN + r] = ostage[2][tid]; // max_3
            out[2 + 3 * NN + r] = ostage[3][tid]; // max_4
            out[2 + 4 * NN + r] = ostage[4][tid]; // max_5
            out[2 + 5 * NN + r] = ostage[5][tid]; // max1_mun
            out[4 + 6 * NN + r] = ostage[6][tid]; // z1
            out[4 + 7 * NN + r] = ostage[6][tid]; // z1 (again)
            out[4 + 8 * NN + r] = ostage[7][tid]; // s
        }
    }
    if (tid == 0) {
        float a = 0.f, b = 0.f, c = 0.f, d = 0.f;
        #pragma unroll
        for (int q = 0; q < NWAVES; ++q) {
            a += pstage[q][0]; b += pstage[q][1];
            c += pstage[q][2]; d += pstage[q][3];
        }
        const int B = nblocks;
        ws[0 * B + blockIdx.x] = a;
        ws[1 * B + blockIdx.x] = b;
        ws[2 * B + blockIdx.x] = c;
        ws[3 * B + blockIdx.x] = d;
    }
}

__global__ void __launch_bounds__(256)
hb_loss_reduce(const float* __restrict__ ws, float* __restrict__ out,
               int B, int N) {
    __shared__ float sm[256][4];
    const int t = threadIdx.x;
    float a = 0.f, b = 0.f, c = 0.f, d = 0.f;
    for (int i = t; i < B; i += 256) {
        a += ws[0 * B + i];
        b += ws[1 * B + i];
        c += ws[2 * B + i];
        d += ws[3 * B + i];
    }
    sm[t][0] = a; sm[t][1] = b; sm[t][2] = c; sm[t][3] = d;
    __syncthreads();
    for (int s = 128; s; s >>= 1) {
        if (t < s) {
            sm[t][0] += sm[t + s][0];
            sm[t][1] += sm[t + s][1];
            sm[t][2] += sm[t + s][2];
            sm[t][3] += sm[t + s][3];
        }
        __syncthreads();
    }
    if (t == 0) {
        const size_t NN = (size_t)N;
        out[0]          = sm[0][0] / (float)N; // loss (mean)
        out[1]          = sm[0][1];            // fy_cs_num
        out[2 + 6 * NN] = sm[0][2];            // tk
        out[3 + 6 * NN] = sm[0][3];            // tk2
    }
}

extern "C" void kernel_launch(void* const* d_in, const int* in_sizes, int n_in,
                              void* d_out, int out_size, void* d_ws, size_t ws_size,
                              hipStream_t stream) {
    const float* y_pred = (const float*)d_in[0];
    const int*   yv     = (const int*)d_in[1];
    const int*   ycs    = (const int*)d_in[2];
    const int*   epoch  = (const int*)d_in[3];
    const int N = in_sizes[1];
    const int nblocks = (N + ROWS_PER_BLOCK - 1) / ROWS_PER_BLOCK;
    // ws usage: 4 arrays of nblocks floats (128KB for N=262144).
    hb_loss_main<<<nblocks, NWAVES * 32, 0, stream>>>(
        y_pred, yv, ycs, epoch, (float*)d_out, (float*)d_ws, N, nblocks);
    hb_loss_reduce<<<1, 256, 0, stream>>>((const float*)d_ws, (float*)d_out,
                                          nblocks, N);
}